// DropBlock_5669356833156
// MI455X (gfx1250) — compile-verified
//
#include <hip/hip_runtime.h>
#include <hip/hip_bf16.h>
#include <cstdint>

// DropBlock: B=64, C=256, H=W=56, mask Hm=Wm=50, block=7, f32 in/out.
// Memory-bound: ~575 MB of HBM traffic -> ~25us floor at 23.3 TB/s.
//   pass 1: per-image TDM DMA of the 50x50 mask tile into LDS (8 wave32s,
//           disjoint row slices), branchless separable 7x7 dilation
//           (row-max then col-max), integer count of dilated ones.
//   pass 2: recompute dilation (mask stays in 192MB L2 because x/out use
//           non-temporal hints), stream x -> out with float4 accesses.

#define HM 50
#define WM 50
#define HH 56
#define WW 56
#define IMG_ELEMS (HH * WW)     // 3136
#define IMG_VEC4 (IMG_ELEMS / 4)  // 784, WW/4 = 14 vec4 per row
#define MASK_ELEMS (HM * WM)    // 2500
#define N_IMG (64 * 256)        // 16384
#define COUNT_M_F 51380224.0f   // 64*256*56*56 (exactly representable in f32)

typedef unsigned int u32x4 __attribute__((ext_vector_type(4)));
typedef int i32x8 __attribute__((ext_vector_type(8)));
typedef int i32x4 __attribute__((ext_vector_type(4)));
typedef float f32x4 __attribute__((ext_vector_type(4)));

#if defined(__gfx1250__) && __has_builtin(__builtin_amdgcn_tensor_load_to_lds) && __has_builtin(__builtin_amdgcn_s_wait_tensorcnt)
#define USE_TDM 1
#endif

__device__ __forceinline__ int clamp49(int j) {
  j = j < 0 ? 0 : j;
  return j > 49 ? 49 : j;
}

#ifdef USE_TDM
// DMA `nrows` rows of 50 f32 from global (row stride 50 elems) into LDS at
// byte offset lds_off. Descriptor per CDNA5 ISA ch. 8 (Tensor DMA D#).
__device__ __forceinline__ void tdm_load_rows(const float* gsrc,
                                              unsigned lds_off, int nrows) {
  unsigned long long ga = (unsigned long long)(uintptr_t)gsrc;
  u32x4 g0;
  g0.x = 1u;                                   // count=1, user mode, no gather
  g0.y = lds_off;                              // lds_addr (bytes)
  g0.z = (unsigned)(ga & 0xffffffffu);         // global_addr[31:0]
  g0.w = (unsigned)((ga >> 32) & 0x01ffffffu)  // global_addr[56:32]
         | (2u << 30);                         // type = 2 ("image")
  i32x8 g1;
  g1[0] = (2 << 16);           // data_size=2 (4 bytes); no pad/iterate/mask
  g1[1] = (WM << 16);          // tensor_dim0 = 50 (low 16 bits at [63:48])
  g1[2] = (nrows << 16);       // tensor_dim1 low16 at [95:80]
  g1[3] = (WM << 16);          // tile_dim0 = 50 at [127:112]
  g1[4] = nrows;               // tile_dim1 = nrows; tile_dim2 = 0
  g1[5] = WM;                  // tensor_dim0_stride = 50 elems (low 32)
  g1[6] = (MASK_ELEMS << 16);  // tensor_dim1_stride = 2500 (low16 at [223:208])
  g1[7] = 0;                   // tensor_dim1_stride high bits
  i32x4 z4 = {0, 0, 0, 0};     // groups 2/3 unused (2-D tile)
  i32x8 z8 = {0, 0, 0, 0, 0, 0, 0, 0};
  __builtin_amdgcn_tensor_load_to_lds(g0, g1, z4, z4, z8, 0);
}
#endif

// Load one 50x50 mask image into smask (LDS) and produce the 1x7 row-dilated
// 50x56 tile in srow (LDS). Leaves the block past a barrier.
__device__ __forceinline__ void load_mask_and_rowdilate(
    const float* __restrict__ mask_img, float* smask, float* srow) {
  const int tid = threadIdx.x;
#ifdef USE_TDM
  // 8 wave32s, disjoint row slices: waves 0,1 take 7 rows, waves 2..7 take 6.
  const int wave = tid >> 5;
  const int start = wave * 6 + (wave < 2 ? wave : 2);
  const int nrows = 6 + (wave < 2 ? 1 : 0);
  const unsigned lds_off =
      (unsigned)(uintptr_t)(const void*)(smask + start * WM);
  tdm_load_rows(mask_img + start * WM, lds_off, nrows);
  __builtin_amdgcn_s_wait_tensorcnt(0);
#else
  for (int i = tid; i < MASK_ELEMS; i += 256) smask[i] = mask_img[i];
#endif
  __syncthreads();
  // Row dilation, branchless: clamping a tap index into [0,49] only
  // duplicates an in-window value, so the max is unchanged.
  for (int idx = tid; idx < HM * WW; idx += 256) {
    const int h = idx / WW;
    const int w = idx - h * WW;
    const int b = w - 6;
    const float* row = smask + h * WM;
    float m = 0.0f;  // mask values are {0,1}
#pragma unroll
    for (int k = 0; k < 7; ++k) m = fmaxf(m, row[clamp49(b + k)]);
    srow[idx] = m;
  }
  __syncthreads();
}

// Column dilation for one float4 (4 consecutive w at row h), branchless.
__device__ __forceinline__ f32x4 coldilate4(const float* srow, int h, int wq) {
  const int b = h - 6;
  f32x4 m = {0.0f, 0.0f, 0.0f, 0.0f};
#pragma unroll
  for (int k = 0; k < 7; ++k) {
    const f32x4 r = *(const f32x4*)(srow + clamp49(b + k) * WW + wq);
    m.x = fmaxf(m.x, r.x);
    m.y = fmaxf(m.y, r.y);
    m.z = fmaxf(m.z, r.z);
    m.w = fmaxf(m.w, r.w);
  }
  return m;
}

__global__ void __launch_bounds__(256)
dropblock_zero_count(unsigned* __restrict__ g_count) {
  if (threadIdx.x == 0) *g_count = 0u;
}

__global__ void __launch_bounds__(256)
dropblock_count(const float* __restrict__ mask, unsigned* __restrict__ g_count) {
  __shared__ __align__(16) float smask[MASK_ELEMS];
  __shared__ __align__(16) float srow[HM * WW];
  __shared__ unsigned scount;
  const int tid = threadIdx.x;
  if (tid == 0) scount = 0u;
  load_mask_and_rowdilate(mask + (size_t)blockIdx.x * MASK_ELEMS, smask, srow);
  unsigned my = 0;
  for (int q = tid; q < IMG_VEC4; q += 256) {
    const int h = q / (WW / 4);
    const int wq = (q - h * (WW / 4)) * 4;
    const f32x4 m = coldilate4(srow, h, wq);
    my += (m.x > 0.5f) + (m.y > 0.5f) + (m.z > 0.5f) + (m.w > 0.5f);
  }
  // wave32 shuffle reduction, then one LDS atomic per wave
#pragma unroll
  for (int off = 16; off > 0; off >>= 1) my += __shfl_down(my, off, 32);
  if ((tid & 31) == 0) atomicAdd(&scount, my);
  __syncthreads();
  if (tid == 0) atomicAdd(g_count, scount);
}

__global__ void __launch_bounds__(256)
dropblock_apply(const float* __restrict__ x, const float* __restrict__ mask,
                const unsigned* __restrict__ g_count, float* __restrict__ out) {
  __shared__ __align__(16) float smask[MASK_ELEMS];
  __shared__ __align__(16) float srow[HM * WW];
  const int tid = threadIdx.x;
  load_mask_and_rowdilate(mask + (size_t)blockIdx.x * MASK_ELEMS, smask, srow);
  // count_ones(block_mask) = countM - dilated_count
  const float scale = COUNT_M_F / (COUNT_M_F - (float)(*g_count));
  const f32x4* __restrict__ x4 =
      (const f32x4*)(x + (size_t)blockIdx.x * IMG_ELEMS);
  f32x4* __restrict__ o4 = (f32x4*)(out + (size_t)blockIdx.x * IMG_ELEMS);
  for (int q = tid; q < IMG_VEC4; q += 256) {
    const int h = q / (WW / 4);
    const int wq = (q - h * (WW / 4)) * 4;
    const f32x4 m = coldilate4(srow, h, wq);
    const f32x4 xv = __builtin_nontemporal_load(x4 + q);  // stream x past L2
    f32x4 r;
    r.x = (m.x > 0.5f) ? 0.0f : xv.x * scale;
    r.y = (m.y > 0.5f) ? 0.0f : xv.y * scale;
    r.z = (m.z > 0.5f) ? 0.0f : xv.z * scale;
    r.w = (m.w > 0.5f) ? 0.0f : xv.w * scale;
    __builtin_nontemporal_store(r, o4 + q);  // stream out past L2
  }
}

extern "C" void kernel_launch(void* const* d_in, const int* in_sizes, int n_in,
                              void* d_out, int out_size, void* d_ws,
                              size_t ws_size, hipStream_t stream) {
  const float* x = (const float*)d_in[0];     // (64,256,56,56) f32
  const float* mask = (const float*)d_in[1];  // (64,256,50,50) f32
  float* out = (float*)d_out;                 // (64,256,56,56) f32
  unsigned* g_count = (unsigned*)d_ws;        // 4 bytes of scratch

  dropblock_zero_count<<<1, 256, 0, stream>>>(g_count);
  dropblock_count<<<N_IMG, 256, 0, stream>>>(mask, g_count);
  dropblock_apply<<<N_IMG, 256, 0, stream>>>(x, mask, g_count, out);
}